// YOLOLoss_85993835201117
// MI455X (gfx1250) — compile-verified
//
#include <hip/hip_runtime.h>
#include <hip/hip_bf16.h>

// Problem constants from the reference
#define BB 16
#define AA 3
#define HH 48
#define WW 96
#define CC 80
#define DD 85
#define TT 256
#define NCELL (BB*AA*HH*WW)   // 221184 = 864 * 256 exactly

typedef __attribute__((ext_vector_type(2))) float v2f;
typedef __attribute__((ext_vector_type(8))) float v8f;

__device__ __forceinline__ float softplusf(float z) {
    // softplus(z) = log(1+exp(z)), numerically stable
    return fmaxf(z, 0.0f) + log1pf(__expf(-fabsf(z)));
}

// Exact 32-lane sum using V_WMMA_F32_16X16X4_F32.
// A[m,k]: lane m (0..15) VGPR0 -> K=0 ; lane m+16 VGPR0 -> K=2 ; VGPR1 lanes -> K=1/3 (zeroed).
// B = all ones (layout-irrelevant). D[m][n] = sum_k A[m,k] = p[m] + p[m+16].
// Per lane, summing its 8 D VGPRs gives sum over 8 rows; xor-16 shuffle adds the other half.
__device__ __forceinline__ float wmma_wave_sum(float p) {
    v2f a; a[0] = p;    a[1] = 0.0f;
    v2f b; b[0] = 1.0f; b[1] = 1.0f;
    v8f c = {};
    v8f d = __builtin_amdgcn_wmma_f32_16x16x4_f32(
        /*neg_a=*/false, a, /*neg_b=*/false, b,
        /*c_mod=*/(short)0, c, /*reuse_a=*/false, /*reuse_b=*/false);
    float s = ((d[0]+d[1])+(d[2]+d[3])) + ((d[4]+d[5])+(d[6]+d[7]));
    s += __shfl_xor(s, 16, 32);
    return s;   // every lane holds the full 32-lane sum
}

__global__ __launch_bounds__(256)
void yolo_clear_kernel(int* __restrict__ cellMap, float* __restrict__ out) {
    int i = blockIdx.x * 256 + threadIdx.x;
    cellMap[i] = 0;
    if (i == 0) out[0] = 0.0f;
}

// One block of 256 threads: per-target anchor matching + scatter (last target wins via atomicMax).
__global__ __launch_bounds__(256)
void yolo_targets_kernel(const float* __restrict__ anchors,
                         const float* __restrict__ txywh,
                         const int*   __restrict__ t_b,
                         int*   __restrict__ cellMap,
                         float4* __restrict__ tgt) {
    int t = threadIdx.x;
    float4 q = ((const float4*)txywh)[t];
    float x = q.x * (float)WW;           // * 96
    float y = q.y * (float)HH;           // * 48
    float w = q.z * 768.0f;              // * INPUT_DIM*2
    float h = q.w * 384.0f;              // * INPUT_DIM
    bool valid = (x >= 0.0f) && (y >= 0.0f) && (x <= (float)(WW-1)) && (y <= (float)(HH-1));
    int gx = min(max((int)floorf(x), 0), WW - 1);
    int gy = min(max((int)floorf(y), 0), HH - 1);

    // argmax IoU over 3 anchors (strict > keeps the first max, like jnp.argmax)
    float bestR = -1.0f; int best = 0; float bsw = 1.0f, bsh = 1.0f;
#pragma unroll
    for (int a = 0; a < AA; ++a) {
        float sw = anchors[2*a + 0] * 8.0f;   // stride_w = 768/96
        float sh = anchors[2*a + 1] * 8.0f;   // stride_h = 384/48
        float inter = fminf(w, sw) * fminf(h, sh);
        float uni   = w*h + sw*sh - inter + 1e-16f;
        float r = inter / uni;
        if (r > bestR) { bestR = r; best = a; bsw = sw; bsh = sh; }
    }

    float4 rec;
    rec.x = x - floorf(x);
    rec.y = y - floorf(y);
    rec.z = logf(w / bsw + 1e-16f);
    rec.w = logf(h / bsh + 1e-16f);
    tgt[t] = rec;

    if (valid) {
        int b = t_b[t];
        int cell = ((b*AA + best)*HH + gy)*WW + gx;
        atomicMax(&cellMap[cell], t + 1);   // highest index wins == scatter "last wins"
    }
}

// Main pass: only channel 4 for the 221k non-object cells; full 85 channels for <=256 target cells.
__global__ __launch_bounds__(256)
void yolo_loss_kernel(const float* __restrict__ pred,
                      const int*   __restrict__ cellMap,
                      const float4* __restrict__ tgt,
                      const int*   __restrict__ t_cls,
                      float* __restrict__ out) {
    const int i = blockIdx.x * 256 + threadIdx.x;   // grid fits exactly -> EXEC all-ones
    const float* p = pred + (size_t)i * DD;
    float z_obj = p[4];
    int tid = cellMap[i] - 1;

    float loss;
    if (tid < 0) {
        loss = 0.5f * softplusf(z_obj);             // no_obj_weight * -log_sigmoid(-z)
    } else {
        loss = softplusf(-z_obj);                   // -log_sigmoid(z)
        float4 tr = tgt[tid];
        float px = 1.0f / (1.0f + __expf(-p[0]));
        float py = 1.0f / (1.0f + __expf(-p[1]));
        float dx = px - tr.x, dy = py - tr.y;
        float dw = p[2] - tr.z, dh = p[3] - tr.w;
        loss += dx*dx + dy*dy + dw*dw + dh*dh;
        // one-hot BCE: sum_c softplus(z_c) - z_{t_cls}
        float cs = 0.0f;
#pragma unroll 4
        for (int c = 0; c < CC; ++c) cs += softplusf(p[5 + c]);
        cs -= p[5 + t_cls[tid]];
        loss += cs;
    }

    // wave reduction via WMMA (branches rejoined; all 32 lanes active)
    float wsum = wmma_wave_sum(loss);

    __shared__ float lsum[8];
    int lane = threadIdx.x & 31;
    int wv   = threadIdx.x >> 5;
    if (lane == 0) lsum[wv] = wsum;
    __syncthreads();
    if (threadIdx.x == 0) {
        float tot = 0.0f;
#pragma unroll
        for (int k = 0; k < 8; ++k) tot += lsum[k];
        atomicAdd(out, tot);
    }
}

extern "C" void kernel_launch(void* const* d_in, const int* in_sizes, int n_in,
                              void* d_out, int out_size, void* d_ws, size_t ws_size,
                              hipStream_t stream) {
    const float* pred    = (const float*)d_in[0];
    const float* anchors = (const float*)d_in[1];
    const float* txywh   = (const float*)d_in[2];
    const int*   t_b     = (const int*)d_in[3];
    const int*   t_cls   = (const int*)d_in[4];
    float* out = (float*)d_out;

    int*    cellMap = (int*)d_ws;                                   // NCELL ints
    float4* tgt     = (float4*)((char*)d_ws + (size_t)NCELL * 4);   // TT float4

    yolo_clear_kernel<<<NCELL/256, 256, 0, stream>>>(cellMap, out);
    yolo_targets_kernel<<<1, TT, 0, stream>>>(anchors, txywh, t_b, cellMap, tgt);
    yolo_loss_kernel<<<NCELL/256, 256, 0, stream>>>(pred, cellMap, tgt, t_cls, out);
}